// DeepJetConstraint_11544872092147
// MI455X (gfx1250) — compile-verified
//
#include <hip/hip_runtime.h>
#include <stdint.h>

// DeepJet constraint transform, MI455X (gfx1250).
// Bandwidth-bound: 256 MB in + 240 MB out @ 23.3 TB/s ~= 21 us floor.
// Strategy: b128 coalesced input loads; LDS-staged output written with the
// gfx1250 async LDS->global engine (global_store_async_from_lds_b128) so the
// 60-byte output rows become fully contiguous, 16B-aligned 128-bit stores.

#define ROWS_PER_BLOCK 256
#define NCOLS_IN  16
#define NCOLS_OUT 15

__device__ __forceinline__ float logit_clipped(float a) {
    // clip to [1e-6, 1-1e-6], then log(a/(1-a)) with fast HW rcp/log
    a = fminf(fmaxf(a, 1e-6f), 1.0f - 1e-6f);
    float r = a * __builtin_amdgcn_rcpf(1.0f - a);
    return __logf(r);
}

__global__ __launch_bounds__(ROWS_PER_BLOCK)
void deepjet_constraint_kernel(const float* __restrict__ x,
                               float* __restrict__ out,
                               int n_rows) {
    // 256 rows * 15 floats = 15360 bytes of staging, 16B aligned.
    __shared__ __align__(16) float out_tile[ROWS_PER_BLOCK * NCOLS_OUT];

    const int tid  = threadIdx.x;
    const int row0 = blockIdx.x * ROWS_PER_BLOCK;
    const int row  = row0 + tid;

    if (row < n_rows) {
        // 64B row, 16B aligned -> 4x global_load_b128, coalesced per wave.
        const float4* rowp = (const float4*)(x + (size_t)row * NCOLS_IN);
        float4 v0 = rowp[0];
        float4 v1 = rowp[1];
        float4 v2 = rowp[2];
        float4 v3 = rowp[3];

        // Stage prefix columns 0..9 (stride-15 LDS writes: odd stride ->
        // bank-conflict-free across a wave32).
        float* o = &out_tile[tid * NCOLS_OUT];
        o[0] = v0.x; o[1] = v0.y; o[2] = v0.z; o[3] = v0.w;
        o[4] = v1.x; o[5] = v1.y; o[6] = v1.z; o[7] = v1.w;
        o[8] = v2.x; o[9] = v2.y;

        // Softmax over columns 10..13 (x normal -> exp safe without max-sub).
        float e0 = __expf(v2.z);   // x[:,10]
        float e1 = __expf(v2.w);   // x[:,11]
        float e2 = __expf(v3.x);   // x[:,12]
        float e3 = __expf(v3.y);   // x[:,13]
        float inv = __builtin_amdgcn_rcpf(e0 + e1 + e2 + e3);
        float s0 = e0 * inv, s1 = e1 * inv, s2 = e2 * inv, s3 = e3 * inv;

        float a0 = s0;
        float a1 = s1;
        float a2 = s1 * __builtin_amdgcn_rcpf(s1 + s0);
        float a3 = s1 * __builtin_amdgcn_rcpf(s1 + s2 + s3);
        float a4 = s3 * __builtin_amdgcn_rcpf(s3 + s2);

        o[10] = logit_clipped(a0);
        o[11] = logit_clipped(a1);
        o[12] = logit_clipped(a2);
        o[13] = logit_clipped(a3);
        o[14] = logit_clipped(a4);
    }
    __syncthreads();

    // --- Async stream-out: LDS tile -> global, contiguous b128 chunks ---
    int valid_rows = n_rows - row0;
    if (valid_rows > ROWS_PER_BLOCK) valid_rows = ROWS_PER_BLOCK;
    if (valid_rows > 0) {
        const int valid_floats = valid_rows * NCOLS_OUT;   // 3840 for full block
        const int full_chunks  = valid_floats >> 2;        // 960 x 16B

        // Low 32 bits of the flat address of a __shared__ object are the
        // workgroup-relative LDS byte offset (flat-LDS aperture mapping).
        const uint32_t lds_base = (uint32_t)(uintptr_t)&out_tile[0];
        // Block's output byte offset: row0*60 < 2^31 for this problem size.
        const uint32_t blk_byte = (uint32_t)((uint64_t)row0 * NCOLS_OUT * sizeof(float));
        const unsigned long long gbase = (unsigned long long)(uintptr_t)out;

        for (int i = tid; i < full_chunks; i += ROWS_PER_BLOCK) {
            uint32_t goff = blk_byte + (uint32_t)i * 16u;   // 16B aligned
            uint32_t loff = lds_base + (uint32_t)i * 16u;   // 16B aligned
            // GVS mode: mem_addr = SGPR64 + VGPR32; VSRC = LDS byte address.
            asm volatile("global_store_async_from_lds_b128 %0, %1, %2"
                         :
                         : "v"(goff), "v"(loff), "s"(gbase)
                         : "memory");
        }

        // Tail floats (only when valid_rows*15 % 4 != 0, i.e. ragged last block)
        const int rem = valid_floats & 3;
        if (rem && tid == 0) {
            size_t gidx = (size_t)row0 * NCOLS_OUT + (size_t)(full_chunks << 2);
            int    lidx = full_chunks << 2;
            for (int k = 0; k < rem; ++k)
                out[gidx + k] = out_tile[lidx + k];
        }
    }
    // S_ENDPGM performs an implicit wait-idle, retiring ASYNCcnt; the LDS
    // tile is not reused, so no explicit s_wait_asynccnt is required.
}

extern "C" void kernel_launch(void* const* d_in, const int* in_sizes, int n_in,
                              void* d_out, int out_size, void* d_ws, size_t ws_size,
                              hipStream_t stream) {
    const float* x  = (const float*)d_in[0];
    float* out      = (float*)d_out;
    const int n_rows = in_sizes[0] / NCOLS_IN;   // 4,000,000
    const int blocks = (n_rows + ROWS_PER_BLOCK - 1) / ROWS_PER_BLOCK;
    deepjet_constraint_kernel<<<blocks, ROWS_PER_BLOCK, 0, stream>>>(x, out, n_rows);
}